// ElectricalNoiseFilter_65111704207643
// MI455X (gfx1250) — compile-verified
//
#include <hip/hip_runtime.h>

// ---------------------------------------------------------------------------
// Cascaded biquad notch filter (scipy sosfilt, DF2T-equivalent) on gfx1250.
//
// Block-parallel IIR: per 16-sample block, zero-state response y0 = H*x with
// H = 16x16 lower-triangular Toeplitz of the section impulse response,
// computed with V_WMMA_F32_16X16X4_F32 (4 chained K=4 steps -> full K=16).
// Cross-block coupling reduces to a 2x2 affine scan on (y[-1], y[-2]);
// boundary-input contributions (x[-1], x[-2]) are folded in via precomputed
// responses g1/g2 so the scan element stays 2-dimensional.
// ---------------------------------------------------------------------------

typedef __attribute__((ext_vector_type(2))) float v2f;
typedef __attribute__((ext_vector_type(8))) float v8f;

#define SAMPLES    160000
#define CHUNK      2048      // samples per LDS chunk = 128 blocks of 16
#define CHUNK_BLKS 128
#define NSEC       3
#define TPB        256       // 8 wave32s; each wave does 16 blocks per WMMA set

__global__ __launch_bounds__(TPB, 1)
void hum_notch_wmma(const float* __restrict__ audio,
                    const float* __restrict__ sos,
                    float* __restrict__ out,
                    int batch)
{
  __shared__ float buf[2][CHUNK];               // ping-pong chunk buffers
  __shared__ float eArr0[CHUNK_BLKS], eArr1[CHUNK_BLKS]; // per-block scan add
  __shared__ float sArr0[CHUNK_BLKS], sArr1[CHUNK_BLKS]; // incoming y-state
  __shared__ float coef[NSEC][5][16];           // 0:himp 1:h1 2:h2 3:g1 4:g2
  __shared__ float carry[NSEC][4];              // y[-1], y[-2], x[-1], x[-2]
  __shared__ float pend[2];

  const int tid  = threadIdx.x;
  const int lane = tid & 31;
  const int wave = tid >> 5;
  const int half = lane >> 4;                   // lane half selects K / M split
  const int m    = lane & 15;                   // A row, B/D column in tile

  const int row = blockIdx.x;
  if (row >= batch) return;
  const float* __restrict__ xrow = audio + (size_t)row * SAMPLES;
  float*       __restrict__ orow = out   + (size_t)row * SAMPLES;

  // ---- per-section response tables (impulse + state/boundary responses) ----
  if (tid < NSEC) {
    const int s = tid;
    const float b0 = sos[s * 6 + 0], b1 = sos[s * 6 + 1], b2 = sos[s * 6 + 2];
    const float a1 = sos[s * 6 + 4], a2 = sos[s * 6 + 5];
    float hi[16], h1[16], h2[16], g1[16], g2[16];
    hi[0] = b0;   hi[1] = b1 - a1 * hi[0];
    h1[0] = -a1;  h1[1] = -a1 * h1[0] - a2;     // response to y[-1] = 1
    h2[0] = -a2;  h2[1] = -a1 * h2[0];          // response to y[-2] = 1
    g1[0] = b1;   g1[1] = b2 - a1 * g1[0];      // response to x[-1] = 1
    g2[0] = b2;   g2[1] = -a1 * g2[0];          // response to x[-2] = 1
    for (int t = 2; t < 16; ++t) {
      hi[t] = ((t == 2) ? b2 : 0.0f) - a1 * hi[t - 1] - a2 * hi[t - 2];
      h1[t] = -a1 * h1[t - 1] - a2 * h1[t - 2];
      h2[t] = -a1 * h2[t - 1] - a2 * h2[t - 2];
      g1[t] = -a1 * g1[t - 1] - a2 * g1[t - 2];
      g2[t] = -a1 * g2[t - 1] - a2 * g2[t - 2];
    }
    for (int t = 0; t < 16; ++t) {
      coef[s][0][t] = hi[t]; coef[s][1][t] = h1[t]; coef[s][2][t] = h2[t];
      coef[s][3][t] = g1[t]; coef[s][4][t] = g2[t];
    }
  }
  if (tid < NSEC * 4) carry[tid >> 2][tid & 3] = 0.0f;
  __syncthreads();

  // ---- hoist constant A fragments: H[m][k] = himp[m-k] (lower triangular) --
  // 16x4 f32 A layout: lanes 0-15 hold K = {k0, k0+1}; lanes 16-31 K = {k0+2, k0+3}
  v2f aReg[NSEC][4];
  for (int s = 0; s < NSEC; ++s) {
    for (int ks = 0; ks < 4; ++ks) {
      const int k = ks * 4 + (half ? 2 : 0);
      v2f a;
      a.x = (m >= k    ) ? coef[s][0][m - k    ] : 0.0f;
      a.y = (m >= k + 1) ? coef[s][0][m - k - 1] : 0.0f;
      aReg[s][ks] = a;
    }
  }

  int cur = 0;
  const int nChunks = (SAMPLES + CHUNK - 1) / CHUNK;
  for (int c = 0; c < nChunks; ++c) {
    const int base   = c * CHUNK;
    const int nValid = min(CHUNK, SAMPLES - base);

    // prefetch next chunk's cachelines while this one is processed
    if (base + CHUNK + tid * 8 < SAMPLES)
      __builtin_prefetch(&xrow[base + CHUNK + tid * 8], 0, 1);

    // stage chunk into LDS (zero-padded tail)
    for (int i = tid; i < CHUNK; i += TPB)
      buf[cur][i] = (base + i < SAMPLES) ? xrow[base + i] : 0.0f;
    __syncthreads();

    for (int s = 0; s < NSEC; ++s) {
      const int y0b   = cur ^ 1;
      const int blk   = wave * 16 + m;     // this lane's block (B/D column)
      const int baseI = blk * 16;

      // ---- Stage A: zero-state response, 16 blocks x 16 samples per wave --
      v8f acc = {};
#pragma unroll
      for (int ks = 0; ks < 4; ++ks) {
        const int k = ks * 4 + (half ? 2 : 0);
        v2f b;
        b.x = buf[cur][baseI + k];
        b.y = buf[cur][baseI + k + 1];
        acc = __builtin_amdgcn_wmma_f32_16x16x4_f32(
            false, aReg[s][ks], false, b, (short)0, acc, false, false);
      }
      // D layout: acc[v] = Y0[v + 8*half][blk]
#pragma unroll
      for (int v = 0; v < 8; ++v)
        buf[y0b][baseI + v + 8 * half] = acc[v];

      // per-block scan additive term e_b (rows 15/14 live in lane half 1)
      if (half) {
        const float xm1 = blk ? buf[cur][baseI - 1] : carry[s][2];
        const float xm2 = blk ? buf[cur][baseI - 2] : carry[s][3];
        eArr0[blk] = acc[7] + coef[s][3][15] * xm1 + coef[s][4][15] * xm2;
        eArr1[blk] = acc[6] + coef[s][3][14] * xm1 + coef[s][4][14] * xm2;
      }
      __syncthreads();

      // ---- Stage B: 2x2 affine scan across blocks (short serial chain) ----
      if (tid == 0) {
        float sy1 = carry[s][0], sy2 = carry[s][1];
        const float M11 = coef[s][1][15], M12 = coef[s][2][15];
        const float M21 = coef[s][1][14], M22 = coef[s][2][14];
        const int nb = nValid >> 4;
        for (int bi = 0; bi < nb; ++bi) {
          sArr0[bi] = sy1; sArr1[bi] = sy2;
          const float ny1 = fmaf(M11, sy1, fmaf(M12, sy2, eArr0[bi]));
          const float ny2 = fmaf(M21, sy1, fmaf(M22, sy2, eArr1[bi]));
          sy1 = ny1; sy2 = ny2;
        }
        pend[0] = sy1; pend[1] = sy2;
      }
      __syncthreads();

      // ---- Stage C: superposition fix-up (fully parallel) -----------------
      for (int i = tid; i < nValid; i += TPB) {
        const int bi = i >> 4, t = i & 15;
        const float xm1 = bi ? buf[cur][bi * 16 - 1] : carry[s][2];
        const float xm2 = bi ? buf[cur][bi * 16 - 2] : carry[s][3];
        buf[y0b][i] = buf[y0b][i]
            + coef[s][1][t] * sArr0[bi] + coef[s][2][t] * sArr1[bi]
            + coef[s][3][t] * xm1       + coef[s][4][t] * xm2;
      }
      __syncthreads();

      if (tid == 0) {     // commit per-section carry for the next chunk
        carry[s][0] = pend[0];
        carry[s][1] = pend[1];
        carry[s][2] = buf[cur][nValid - 1];
        carry[s][3] = buf[cur][nValid - 2];
      }
      __syncthreads();
      cur = y0b;          // section output becomes next section's input
    }

    // write filtered chunk
    for (int i = tid; i < nValid; i += TPB)
      orow[base + i] = buf[cur][i];
    __syncthreads();
  }
}

extern "C" void kernel_launch(void* const* d_in, const int* in_sizes, int n_in,
                              void* d_out, int out_size, void* d_ws, size_t ws_size,
                              hipStream_t stream) {
  (void)n_in; (void)out_size; (void)d_ws; (void)ws_size;
  const float* audio = (const float*)d_in[0];
  const float* sos   = (const float*)d_in[1];
  float* out = (float*)d_out;
  const int batch = in_sizes[0] / SAMPLES;
  hum_notch_wmma<<<batch, TPB, 0, stream>>>(audio, sos, out, batch);
}